// SourceModel_9122510536838
// MI455X (gfx1250) — compile-verified
//
#include <hip/hip_runtime.h>
#include <math.h>

#define N_NODES 100000
#define N_EDGES 800000
#define LDIM 64
#define MDIM 128
#define UDIM 640
#define NEG_SLOPE 0.01f
#define BN_EPS 1e-5f

typedef __attribute__((ext_vector_type(2))) float v2f;
typedef __attribute__((ext_vector_type(8))) float v8f;

// D = A(16x4) * B(4x16) + C(16x16), all f32, wave32.
__device__ __forceinline__ v8f wmma_f32(v2f a, v2f b, v8f c) {
  return __builtin_amdgcn_wmma_f32_16x16x4_f32(false, a, false, b, (short)0, c,
                                               false, false);
}

__device__ __forceinline__ float lrelu(float v) {
  return v > 0.f ? v : v * NEG_SLOPE;
}

// ---------------------------------------------------------------------------
// Repack a row-major weight W[OUT][IN] into per-lane WMMA B fragments:
// frag[(jt*(IN/4) + k)*32 + lane] = { W[jt*16+lo][4k+2hi], W[...][4k+2hi+1] }
// ---------------------------------------------------------------------------
__global__ void repack_b(const float* __restrict__ W, float* __restrict__ out,
                         int OUT, int IN) {
  const int ksteps = IN >> 2;
  const int total = (OUT >> 4) * ksteps * 32;
  for (int id = blockIdx.x * blockDim.x + threadIdx.x; id < total;
       id += gridDim.x * blockDim.x) {
    const int lane = id & 31;
    const int rest = id >> 5;
    const int k = rest % ksteps;
    const int jt = rest / ksteps;
    const int lo = lane & 15, hi = lane >> 4;
    const int n = jt * 16 + lo;
    const int col = 4 * k + 2 * hi;
    v2f v;
    v.x = W[(long long)n * IN + col];
    v.y = W[(long long)n * IN + col + 1];
    ((v2f*)out)[id] = v;
  }
}

// ---------------------------------------------------------------------------
// Kernel 1: per-edge message MLP (two 128x128 layers via WMMA) + atomic
// scatter of raw moments m, m^2, m^3, m^4 and edge count per source node.
// One wave = one 16-edge tile. 4 waves / block.
// ---------------------------------------------------------------------------
#define K1_WAVES 4
#define TSTRIDE 132  // 16 rows x 132 floats: 8B-aligned rows, conflict-free

__global__ __launch_bounds__(128) void edge_msg_moments(
    const float* __restrict__ x_t, const float* __restrict__ eattr,
    const long long* __restrict__ ei, const float* __restrict__ W1r,
    const float* __restrict__ b1, const float* __restrict__ W2r,
    const float* __restrict__ b2, float* __restrict__ moments,
    float* __restrict__ cnt) {
  __shared__ float tst_all[K1_WAVES][16 * TSTRIDE];
  const int wave = threadIdx.x >> 5;
  const int lane = threadIdx.x & 31;
  const int lo = lane & 15, hi = lane >> 4;
  float* tst = tst_all[wave];

  const int tile = blockIdx.x * K1_WAVES + wave;
  const int e0 = tile * 16;

  const long long tgt = ei[N_EDGES + e0 + lo];
  const long long srcl = ei[e0 + lo];
  if (hi == 0) atomicAdd(&cnt[srcl], 1.0f);

  long long se[8];
#pragma unroll
  for (int r = 0; r < 8; ++r) se[r] = ei[e0 + r + 8 * hi];

  // A fragments for layer 1: K=128 = concat(x_t[tgt] (64), edge_attr (64))
  v2f a1[32];
  const float* xr = x_t + tgt * LDIM;
  const float* ar = eattr + (long long)(e0 + lo) * LDIM;
#pragma unroll
  for (int k = 0; k < 16; ++k) a1[k] = *(const v2f*)(xr + 4 * k + 2 * hi);
#pragma unroll
  for (int k = 0; k < 16; ++k) a1[16 + k] = *(const v2f*)(ar + 4 * k + 2 * hi);

  // Layer 1: t = lrelu(msg_in @ W1^T + b1), staged to LDS for transpose.
  for (int jt = 0; jt < 8; ++jt) {
    const float bias = b1[jt * 16 + lo];
    v8f c = {bias, bias, bias, bias, bias, bias, bias, bias};
    const v2f* bp = (const v2f*)W1r + (jt * 32) * 32 + lane;
#pragma unroll
    for (int k = 0; k < 32; ++k) c = wmma_f32(a1[k], bp[k * 32], c);
#pragma unroll
    for (int r = 0; r < 8; ++r)
      tst[(r + 8 * hi) * TSTRIDE + jt * 16 + lo] = lrelu(c[r]);
  }
  __syncthreads();

  // Reload hidden in A-fragment layout (transpose through LDS).
  v2f a2[32];
#pragma unroll
  for (int k = 0; k < 32; ++k)
    a2[k] = *(const v2f*)(tst + lo * TSTRIDE + 4 * k + 2 * hi);

  // Layer 2: msg = t @ W2^T + b2, then scatter raw moments by src.
  for (int jt = 0; jt < 8; ++jt) {
    const float bias = b2[jt * 16 + lo];
    v8f c = {bias, bias, bias, bias, bias, bias, bias, bias};
    const v2f* bp = (const v2f*)W2r + (jt * 32) * 32 + lane;
#pragma unroll
    for (int k = 0; k < 32; ++k) c = wmma_f32(a2[k], bp[k * 32], c);
#pragma unroll
    for (int r = 0; r < 8; ++r) {
      const float v = c[r];
      const float v2 = v * v;
      float* base = moments + se[r] * 512 + jt * 16 + lo;
      atomicAdd(base, v);
      atomicAdd(base + 128, v2);
      atomicAdd(base + 256, v2 * v);
      atomicAdd(base + 384, v2 * v2);
    }
  }
}

// ---------------------------------------------------------------------------
// Kernel 2: convert raw moments -> (mean, std, skew, kurt) in place.
// ---------------------------------------------------------------------------
__global__ void node_stats(float* __restrict__ moments,
                           const float* __restrict__ cnt) {
  const int id = blockIdx.x * blockDim.x + threadIdx.x;
  if (id >= N_NODES * MDIM) return;
  const int i = id >> 7;     // /128
  const int j = id & 127;    // %128
  float* p = moments + (long long)i * 512 + j;
  const float denom = fmaxf(cnt[i], 1.0f);
  const float inv = 1.0f / denom;
  const float m1 = p[0] * inv;
  const float m2 = p[128] * inv;
  const float m3 = p[256] * inv;
  const float m4 = p[384] * inv;
  float var = m2 - m1 * m1;
  var = lrelu(var);
  const float std = sqrtf(var + 1e-6f);
  const float m1sq = m1 * m1;
  const float m3c = m3 - 3.f * m1 * m2 + 2.f * m1 * m1sq;
  const float m4c = m4 - 4.f * m1 * m3 + 6.f * m1sq * m2 - 3.f * m1sq * m1sq;
  const float is = 1.0f / std;
  float skew = m3c * is * is * is;
  float kurt = m4c * is * is * is * is;
  if (skew != skew) skew = 0.f;
  if (kurt != kurt) kurt = 0.f;
  p[0] = m1;
  p[128] = std;
  p[256] = skew;
  p[384] = kurt;
}

// ---------------------------------------------------------------------------
// Kernel 3: update MLP. One 16-node tile per 8-wave workgroup.
// feat [16,640] and hidden [16,640] staged in dynamic LDS.
// Layer 1 (640x640): 40 jt tiles, 5 per wave. Layer 2 (64x640): waves 0-3.
// ---------------------------------------------------------------------------
#define FS 644  // padded feature stride

__global__ __launch_bounds__(256) void update_mlp(
    const float* __restrict__ x_s, const float* __restrict__ x_u,
    const float* __restrict__ stats, const float* __restrict__ uW1r,
    const float* __restrict__ ub1, const float* __restrict__ uW2r,
    const float* __restrict__ ub2, float* __restrict__ h) {
  extern __shared__ float smem[];
  float* sfeat = smem;           // 16*FS
  float* sh1 = smem + 16 * FS;   // 16*FS
  const int tid = threadIdx.x;
  const int wave = tid >> 5, lane = tid & 31;
  const int lo = lane & 15, hi = lane >> 4;
  const long long n0 = (long long)blockIdx.x * 16;

  // Cooperative feature build: [x_s | mean | std | skew | kurt | x_u]
  for (int idx = tid; idx < 16 * UDIM; idx += 256) {
    const int row = idx / UDIM, col = idx % UDIM;
    const long long i = n0 + row;
    float v;
    if (col < 64)
      v = x_s[i * 64 + col];
    else if (col < 576)
      v = stats[i * 512 + (col - 64)];
    else
      v = x_u[col - 576];
    sfeat[row * FS + col] = v;
  }
  __syncthreads();

  // Layer 1: h1 = lrelu(feat @ uW1^T + ub1)
  for (int jtl = 0; jtl < 5; ++jtl) {
    const int jt = wave * 5 + jtl;
    const float bias = ub1[jt * 16 + lo];
    v8f c = {bias, bias, bias, bias, bias, bias, bias, bias};
    const v2f* bp = (const v2f*)uW1r + (jt * 160) * 32 + lane;
#pragma unroll 8
    for (int k = 0; k < 160; ++k) {
      const v2f a = *(const v2f*)(sfeat + lo * FS + 4 * k + 2 * hi);
      c = wmma_f32(a, bp[k * 32], c);
    }
#pragma unroll
    for (int r = 0; r < 8; ++r)
      sh1[(r + 8 * hi) * FS + jt * 16 + lo] = lrelu(c[r]);
  }
  __syncthreads();

  // Layer 2: h = h1 @ uW2^T + ub2 (4 jt tiles on waves 0-3; wave-uniform)
  if (wave < 4) {
    const int jt = wave;
    const float bias = ub2[jt * 16 + lo];
    v8f c = {bias, bias, bias, bias, bias, bias, bias, bias};
    const v2f* bp = (const v2f*)uW2r + (jt * 160) * 32 + lane;
#pragma unroll 8
    for (int k = 0; k < 160; ++k) {
      const v2f a = *(const v2f*)(sh1 + lo * FS + 4 * k + 2 * hi);
      c = wmma_f32(a, bp[k * 32], c);
    }
#pragma unroll
    for (int r = 0; r < 8; ++r)
      h[(n0 + r + 8 * hi) * 64 + jt * 16 + lo] = c[r];
  }
}

// ---------------------------------------------------------------------------
// BatchNorm (training-mode forward, biased variance)
// ---------------------------------------------------------------------------
__global__ void bn_reduce(const float* __restrict__ h, float* __restrict__ bns) {
  const int tid = blockIdx.x * blockDim.x + threadIdx.x;
  const int j = tid & 63;
  const int stride = (gridDim.x * blockDim.x) >> 6;
  float s = 0.f, q = 0.f;
  for (int row = tid >> 6; row < N_NODES; row += stride) {
    const float v = h[(long long)row * 64 + j];
    s += v;
    q += v * v;
  }
  atomicAdd(&bns[j], s);
  atomicAdd(&bns[64 + j], q);
}

__global__ void bn_apply(const float* __restrict__ h,
                         const float* __restrict__ bns,
                         const float* __restrict__ gamma,
                         const float* __restrict__ beta,
                         float* __restrict__ out) {
  const long long id = (long long)blockIdx.x * blockDim.x + threadIdx.x;
  if (id >= (long long)N_NODES * 64) return;
  const int j = (int)(id & 63);
  const float mu = bns[j] * (1.0f / N_NODES);
  const float var = bns[64 + j] * (1.0f / N_NODES) - mu * mu;
  out[id] = gamma[j] * (h[id] - mu) * rsqrtf(var + BN_EPS) + beta[j];
}

// ---------------------------------------------------------------------------
extern "C" void kernel_launch(void* const* d_in, const int* in_sizes, int n_in,
                              void* d_out, int out_size, void* d_ws,
                              size_t ws_size, hipStream_t stream) {
  const float* x_s = (const float*)d_in[0];
  const float* x_t = (const float*)d_in[1];
  const float* eattr = (const float*)d_in[2];
  const float* x_u = (const float*)d_in[3];
  const float* mW1 = (const float*)d_in[4];
  const float* mb1 = (const float*)d_in[5];
  const float* mW2 = (const float*)d_in[6];
  const float* mb2 = (const float*)d_in[7];
  const float* uW1 = (const float*)d_in[8];
  const float* ub1 = (const float*)d_in[9];
  const float* uW2 = (const float*)d_in[10];
  const float* ub2 = (const float*)d_in[11];
  const float* gamma = (const float*)d_in[12];
  const float* beta = (const float*)d_in[13];
  const long long* ei = (const long long*)d_in[14];

  char* ws = (char*)d_ws;
  const size_t OFF_MOM = 0;
  const size_t OFF_CNT = (size_t)N_NODES * 512 * 4;        // 204,800,000
  const size_t OFF_W1R = OFF_CNT + (size_t)N_NODES * 4;    // 205,200,000
  const size_t OFF_W2R = OFF_W1R + 8 * 32 * 32 * 8;        // +64KB
  const size_t OFF_UW1R = OFF_W2R + 8 * 32 * 32 * 8;       // +64KB
  const size_t OFF_UW2R = OFF_UW1R + (size_t)40 * 160 * 32 * 8;
  const size_t OFF_H = OFF_UW2R + (size_t)4 * 160 * 32 * 8;
  const size_t OFF_BN = OFF_H + (size_t)N_NODES * 64 * 4;

  float* moments = (float*)(ws + OFF_MOM);
  float* cnt = (float*)(ws + OFF_CNT);
  float* W1r = (float*)(ws + OFF_W1R);
  float* W2r = (float*)(ws + OFF_W2R);
  float* uW1r = (float*)(ws + OFF_UW1R);
  float* uW2r = (float*)(ws + OFF_UW2R);
  float* hbuf = (float*)(ws + OFF_H);
  float* bns = (float*)(ws + OFF_BN);

  hipMemsetAsync(ws + OFF_MOM, 0, OFF_W1R, stream);  // moments + counts
  hipMemsetAsync(ws + OFF_BN, 0, 512, stream);       // BN accumulators

  repack_b<<<64, 256, 0, stream>>>(mW1, W1r, MDIM, MDIM);
  repack_b<<<64, 256, 0, stream>>>(mW2, W2r, MDIM, MDIM);
  repack_b<<<512, 256, 0, stream>>>(uW1, uW1r, UDIM, UDIM);
  repack_b<<<64, 256, 0, stream>>>(uW2, uW2r, LDIM, UDIM);

  edge_msg_moments<<<N_EDGES / (16 * K1_WAVES), 128, 0, stream>>>(
      x_t, eattr, ei, W1r, mb1, W2r, mb2, moments, cnt);

  node_stats<<<(N_NODES * MDIM) / 256, 256, 0, stream>>>(moments, cnt);

  update_mlp<<<N_NODES / 16, 256, 2 * 16 * FS * sizeof(float), stream>>>(
      x_s, x_u, moments, uW1r, ub1, uW2r, ub2, hbuf);

  bn_reduce<<<256, 256, 0, stream>>>(hbuf, bns);
  bn_apply<<<(N_NODES * 64 + 255) / 256, 256, 0, stream>>>(hbuf, bns, gamma,
                                                           beta,
                                                           (float*)d_out);
}